// GIN_Link_27152783245329
// MI455X (gfx1250) — compile-verified
//
#include <hip/hip_runtime.h>

#define N_NODES 100000
#define N_EDGES 3200000
#define N_PAIRS 500000
#define D       256      // D_IN == D_HID
#define EPB     16       // edges per scatter block
#define NT      4        // N column-tiles (of 16) per wave -> wave owns 16x64 of C

typedef __attribute__((ext_vector_type(2))) float v2f;
typedef __attribute__((ext_vector_type(8))) float v8f;

// ---------------------------------------------------------------- copy (agg = src)
__global__ __launch_bounds__(256)
void copy_f4(const float4* __restrict__ src, float4* __restrict__ dst, int n4) {
    int i = blockIdx.x * 256 + threadIdx.x;
    if (i < n4) dst[i] = src[i];
}

// ---------------------------------------------------------------- edge scatter-add
// block = 256 threads = one full feature row; EPB edges per block.
// gather x[src] row (coalesced), global_atomic_add_f32 into agg[dst] row.
// Both tables are ~102MB -> L2-resident on the 192MB L2.
__global__ __launch_bounds__(256)
void scatter_add(const float* __restrict__ x, const int* __restrict__ ei,
                 float* __restrict__ agg, int nE) {
    const int f    = threadIdx.x;          // feature 0..255
    const int base = blockIdx.x * EPB;
    #pragma unroll
    for (int i = 0; i < EPB; ++i) {
        int e = base + i;
        if (e < nE) {
            int s = ei[e];                 // src row  (edge_index[0][e])
            int d = ei[nE + e];            // dst row  (edge_index[1][e])
            atomicAdd(&agg[(size_t)d * D + f], x[(size_t)s * D + f]);
        }
    }
}

// ---------------------------------------------------------------- fused GEMM+bias+ReLU
// Out[m][n] = relu( sum_k A[m][k] * W[n][k] + bias[n] ),  A:[M,256] W:[256,256]
// block = 256 thr (8 waves). Block tile: 128(M) x 64(N), K blocked by 16 via LDS.
// Each wave owns a 16x64 strip of C (4 accumulators); per K=4 micro-step the A
// fragment is reused across 4 v_wmma_f32_16x16x4_f32 issues. A and B fragments
// are both K-adjacent float pairs in LDS -> single 8B ds loads into even VGPR
// pairs (no repack movs). Out-of-range M rows are CLAMPED on load (a C row
// depends only on its own A row, so clamped garbage only reaches C rows the
// guarded store discards) -- EXEC stays all-1s through the hot loop.
__global__ __launch_bounds__(256)
void gemm_bias_relu(const float* __restrict__ A, const float* __restrict__ W,
                    const float* __restrict__ bias, float* __restrict__ Out, int M) {
    // row pad = 20 floats: rows stay 16B-aligned (float4/v2f accesses) and
    // column-of-rows reads with row-stride 20 dwords hit 16 distinct LDS banks.
    __shared__ float As[128][20];          // As[m][k]
    __shared__ float Bs[64][20];           // Bs[n][k]  (transposed W tile)

    const int tid  = threadIdx.x;
    const int lane = tid & 31;
    const int wave = tid >> 5;       // 0..7  -> M sub-tile
    const int half = lane >> 4;      // 0/1   -> K half of A/B frag, M half of C
    const int lm   = lane & 15;
    const int m0   = blockIdx.y * 128;
    const int n0   = blockIdx.x * (16 * NT);

    v8f acc[NT] = {};

    for (int kb = 0; kb < D; kb += 16) {
        // ---- stage A: 128x16 tile, 2 float4 per thread, coalesced in k, clamped rows
        #pragma unroll
        for (int i = 0; i < 2; ++i) {
            int q   = tid + i * 256;               // 0..511 float4 slots
            int r   = q >> 2;                      // tile row 0..127
            int c   = (q & 3) * 4;                 // k within tile
            int row = m0 + r;
            row = row < M ? row : (M - 1);         // clamp: branch-free, EXEC stays full
            float4 v = *(const float4*)(A + (size_t)row * D + kb + c);
            *(float4*)&As[r][c] = v;
        }
        // ---- stage B (transposed): Bs[n][k] = W[(n0+n)*D + kb + k]
        {
            int k  = tid & 15;
            int nb = tid >> 4;                     // 0..15
            #pragma unroll
            for (int j = 0; j < NT; ++j)
                Bs[nb + 16 * j][k] = W[(size_t)(n0 + nb + 16 * j) * D + kb + k];
        }
        __syncthreads();

        // ---- 4 WMMA micro-steps of K=4; A frag reused across NT column tiles
        #pragma unroll
        for (int kk = 0; kk < 4; ++kk) {
            int kc = kk * 4 + 2 * half;            // frag layout: v0 K=kc, v1 K=kc+1
            v2f a = *(const v2f*)&As[wave * 16 + lm][kc];   // 8B-aligned pair
            #pragma unroll
            for (int ct = 0; ct < NT; ++ct) {
                v2f b = *(const v2f*)&Bs[ct * 16 + lm][kc]; // 8B-aligned pair
                acc[ct] = __builtin_amdgcn_wmma_f32_16x16x4_f32(
                    false, a, false, b, (short)0, acc[ct], false, false);
            }
        }
        __syncthreads();
    }

    // ---- epilogue: C/D layout row m = v + 8*half, col n = lm
    #pragma unroll
    for (int ct = 0; ct < NT; ++ct) {
        float bn = bias[n0 + ct * 16 + lm];
        #pragma unroll
        for (int v = 0; v < 8; ++v) {
            int m = m0 + wave * 16 + v + 8 * half;
            if (m < M) {
                float r = acc[ct][v] + bn;
                Out[(size_t)m * D + n0 + ct * 16 + lm] = r > 0.f ? r : 0.f;
            }
        }
    }
}

// ---------------------------------------------------------------- pair head
// one wave32 per pair: logits = concat(h[ia],h[ib]) @ W3^T + b3 ; log_softmax
__global__ __launch_bounds__(256)
void pair_head(const float* __restrict__ h, const int* __restrict__ idx,
               const float* __restrict__ W3, const float* __restrict__ b3,
               float* __restrict__ out, int P) {
    int gid  = blockIdx.x * 256 + threadIdx.x;
    int p    = gid >> 5;
    int lane = threadIdx.x & 31;
    if (p >= P) return;

    int ia = idx[2 * p];
    int ib = idx[2 * p + 1];
    const float* ha = h + (size_t)ia * D;
    const float* hb = h + (size_t)ib * D;

    float a0 = 0.f, a1 = 0.f;
    #pragma unroll
    for (int k = lane; k < D; k += 32) {          // 8 iterations, coalesced
        float va = ha[k], vb = hb[k];
        a0 = fmaf(va, W3[k],         fmaf(vb, W3[D + k],     a0));
        a1 = fmaf(va, W3[2 * D + k], fmaf(vb, W3[3 * D + k], a1));
    }
    #pragma unroll
    for (int off = 16; off > 0; off >>= 1) {      // wave32 tree reduce
        a0 += __shfl_xor(a0, off, 32);
        a1 += __shfl_xor(a1, off, 32);
    }
    if (lane == 0) {
        float l0 = a0 + b3[0], l1 = a1 + b3[1];
        float m  = fmaxf(l0, l1);
        float lse = m + logf(expf(l0 - m) + expf(l1 - m));
        out[2 * p]     = l0 - lse;
        out[2 * p + 1] = l1 - lse;
    }
}

// ---------------------------------------------------------------- launcher
extern "C" void kernel_launch(void* const* d_in, const int* in_sizes, int n_in,
                              void* d_out, int out_size, void* d_ws, size_t ws_size,
                              hipStream_t stream) {
    const float* x    = (const float*)d_in[0];
    const int*   ei   = (const int*)  d_in[1];
    const int*   pidx = (const int*)  d_in[2];
    const float* W1   = (const float*)d_in[3];
    const float* b1   = (const float*)d_in[4];
    const float* W2   = (const float*)d_in[5];
    const float* b2   = (const float*)d_in[6];
    const float* W3   = (const float*)d_in[7];
    const float* b3   = (const float*)d_in[8];
    float*       out  = (float*)d_out;

    const size_t nfeat = (size_t)N_NODES * D;     // 25.6M floats per buffer
    float* agg = (float*)d_ws;
    float* h1  = agg + nfeat;
    float* h2  = h1  + nfeat;

    const int  n4 = (int)(nfeat / 4);
    dim3 cgrd((n4 + 255) / 256);
    dim3 sgrd((N_EDGES + EPB - 1) / EPB);
    dim3 ggrd(D / (16 * NT), (N_NODES + 127) / 128);
    dim3 pgrd(((size_t)N_PAIRS * 32 + 255) / 256);

    // ---- layer 1: h1 = relu((x + seg_sum(x)) @ W1^T + b1)
    copy_f4       <<<cgrd, 256, 0, stream>>>((const float4*)x, (float4*)agg, n4);
    scatter_add   <<<sgrd, 256, 0, stream>>>(x, ei, agg, N_EDGES);
    gemm_bias_relu<<<ggrd, 256, 0, stream>>>(agg, W1, b1, h1, N_NODES);

    // ---- layer 2: h2 = relu((h1 + seg_sum(h1)) @ W2^T + b2)
    copy_f4       <<<cgrd, 256, 0, stream>>>((const float4*)h1, (float4*)agg, n4);
    scatter_add   <<<sgrd, 256, 0, stream>>>(h1, ei, agg, N_EDGES);
    gemm_bias_relu<<<ggrd, 256, 0, stream>>>(agg, W2, b2, h2, N_NODES);

    // ---- pair head + log_softmax
    pair_head     <<<pgrd, 256, 0, stream>>>(h2, pidx, W3, b3, out, N_PAIRS);
}